// GPT_45707041964786
// MI455X (gfx1250) — compile-verified
//
#include <hip/hip_runtime.h>

// CDNA5 / gfx1250 WMMA operand vectors
typedef __attribute__((ext_vector_type(2))) float v2f;   // 16x4 f32 A/B operand
typedef __attribute__((ext_vector_type(8))) float v8f;   // 16x16 f32 C/D operand
typedef __attribute__((ext_vector_type(2))) float f32x2; // packed-FMA pair (v_pk_fma_f32)

// Hardware tanh (gfx1250 TRANS op v_tanh_f32).
__device__ __forceinline__ float fast_tanh(float x) {
#if __has_builtin(__builtin_amdgcn_tanhf)
  return __builtin_amdgcn_tanhf(x);
#else
  float y;
  asm("v_tanh_f32 %0, %1" : "=v"(y) : "v"(x));
  return y;
#endif
}

__device__ __forceinline__ f32x2 pk_fma(f32x2 a, f32x2 b, f32x2 c) {
  return __builtin_elementwise_fma(a, b, c);   // -> v_pk_fma_f32 on CDNA5
}

// out[k][s] = sum_n tanh( sum_d x[s][d]*W[n][k][d] + b[n][k] ) * c_k[n]
//
// Per 16-sample tile: D = A x B via V_WMMA_F32_16X16X4_F32 where A (16x4) holds
// 16 permuted channels (n,k) of [W(:,0..2), bias] (bias rides in the K=3 pad)
// and B (4x16) holds [x0;x1;x2;1] for 16 samples. Channels are grouped so tile
// t covers head k = t/4 only. In the D layout each lane owns ONE sample column
// and 8 channel rows, so tanh + packed weight-FMA is entirely per-lane; one
// xor-16 shuffle merges the two row halves at the end.
__global__ __launch_bounds__(256) void GPT_pinn_head_kernel(
    const float* __restrict__ x, const float* __restrict__ W,
    const float* __restrict__ b, const float* __restrict__ c_rho,
    const float* __restrict__ c_p, const float* __restrict__ c_u,
    float* __restrict__ out, int Bn, int numTiles)
{
  const int lane = threadIdx.x & 31;
  const int li   = lane & 15;   // sample column within tile / A row M%16
  const int half = lane >> 4;   // 0: rows j / K=0,1 ; 1: rows j+8 / K=2,3
  const int wavesPerBlock = blockDim.x >> 5;
  const int waveId   = blockIdx.x * wavesPerBlock + (threadIdx.x >> 5);
  const int numWaves = gridDim.x * wavesPerBlock;

  // ---- Loop-invariant A tiles (weights+bias) and packed reduction weights ----
  // A_t layout (32-bit 16x4): lanes 0-15: {K=0,K=1}; lanes 16-31: {K=2,K=3}.
  v2f Amat[12];
  f32x2 wp[12][4];                        // weight pairs for rows {2j, 2j+1}
#pragma unroll
  for (int t = 0; t < 12; ++t) {
    const int k = t >> 2;                 // output head: 0=rho,1=p,2=u
    const int n = ((t & 3) << 4) + li;    // neuron index for this lane's A row
    const float* a0p = half ? (W + n * 9 + k * 3 + 2) : (W + n * 9 + k * 3 + 0);
    const float* a1p = half ? (b + n * 3 + k)         : (W + n * 9 + k * 3 + 1);
    Amat[t].x = *a0p;
    Amat[t].y = *a1p;                     // bias rides in K=3 (B row K=3 is 1.0)
    const float* ck = (k == 0) ? c_rho : ((k == 1) ? c_p : c_u);
    const int base = ((t & 3) << 4) + (half << 3);  // rows j / j+8 per half
#pragma unroll
    for (int jj = 0; jj < 4; ++jj) {
      wp[t][jj].x = ck[base + 2 * jj + 0];
      wp[t][jj].y = ck[base + 2 * jj + 1];
    }
  }

  const int xoff = half << 1;  // lane's element for B row 0/2: x[3s+0] or x[3s+2]

  // ---- Software-pipelined grid-stride loop over 16-sample tiles ----
  int tile = waveId;
  if (tile >= numTiles) return;

  int s0  = tile * 16 + li;
  int ss0 = (s0 < Bn) ? s0 : (Bn - 1);
  float bx = x[ss0 * 3 + xoff];
  float by = x[ss0 * 3 + 1];

  for (; tile < numTiles; tile += numWaves) {
    const int s = tile * 16 + li;
    const float cur_bx = bx, cur_by = by;

    // Prefetch next tile's x (clamped -> unconditional, latency hidden).
    {
      const int nt = tile + numWaves;
      int sn = nt * 16 + li;
      sn = (sn < Bn) ? sn : (Bn - 1);
      bx = x[sn * 3 + xoff];
      by = x[sn * 3 + 1];
    }

    // B operand (4x16): lanes 0-15: {x0, x1}; lanes 16-31: {x2, 1.0}
    v2f Bv;
    Bv.x = cur_bx;
    Bv.y = half ? 1.0f : cur_by;   // one v_cndmask

    // Two packed pair-accumulators per head (even/odd) -> chains <= 8 deep.
    f32x2 accA[3], accB[3];
#pragma unroll
    for (int k = 0; k < 3; ++k) { accA[k] = (f32x2)(0.0f); accB[k] = (f32x2)(0.0f); }

#pragma unroll
    for (int t = 0; t < 12; ++t) {
      const int k = t >> 2;
      v8f cz = {};
      // (neg_a, A, neg_b, B, c_mod, C, reuse_a, reuse_b)
      v8f d = __builtin_amdgcn_wmma_f32_16x16x4_f32(
          false, Amat[t], false, Bv, (short)0, cz, false, false);
      f32x2 t01; t01.x = fast_tanh(d[0]); t01.y = fast_tanh(d[1]);
      f32x2 t23; t23.x = fast_tanh(d[2]); t23.y = fast_tanh(d[3]);
      f32x2 t45; t45.x = fast_tanh(d[4]); t45.y = fast_tanh(d[5]);
      f32x2 t67; t67.x = fast_tanh(d[6]); t67.y = fast_tanh(d[7]);
      accA[k] = pk_fma(t01, wp[t][0], accA[k]);
      accB[k] = pk_fma(t23, wp[t][1], accB[k]);
      accA[k] = pk_fma(t45, wp[t][2], accA[k]);
      accB[k] = pk_fma(t67, wp[t][3], accB[k]);
    }

    float acc[3];
#pragma unroll
    for (int k = 0; k < 3; ++k) {
      const f32x2 sum = accA[k] + accB[k];  // v_pk_add_f32
      acc[k] = sum.x + sum.y;
    }

    // Merge row halves: lane l (rows 0..7) + lane l+16 (rows 8..15), same sample.
    acc[0] += __shfl_xor(acc[0], 16, 32);
    acc[1] += __shfl_xor(acc[1], 16, 32);
    acc[2] += __shfl_xor(acc[2], 16, 32);

    if (half == 0 && s < Bn) {
      out[s]          = acc[0];  // out_rho
      out[Bn + s]     = acc[1];  // out_p
      out[2 * Bn + s] = acc[2];  // out_u
    }
  }
}

extern "C" void kernel_launch(void* const* d_in, const int* in_sizes, int n_in,
                              void* d_out, int out_size, void* d_ws, size_t ws_size,
                              hipStream_t stream) {
  const float* x     = (const float*)d_in[0];
  const float* W     = (const float*)d_in[1];
  const float* b     = (const float*)d_in[2];
  const float* c_rho = (const float*)d_in[3];
  const float* c_p   = (const float*)d_in[4];
  const float* c_u   = (const float*)d_in[5];
  float* out = (float*)d_out;

  const int Bn = in_sizes[0] / 3;          // number of samples
  if (Bn <= 0) return;
  const int numTiles = (Bn + 15) / 16;     // 62500 for B=1e6

  // 256 threads = 8 wave32s per block; grid-stride so the preload amortizes.
  int blocks = 1024;
  const int blocksNeeded = (numTiles + 7) / 8;
  if (blocks > blocksNeeded) blocks = blocksNeeded;

  GPT_pinn_head_kernel<<<blocks, 256, 0, stream>>>(
      x, W, b, c_rho, c_p, c_u, out, Bn, numTiles);
}